// AdditiveAttention_13795434955425
// MI455X (gfx1250) — compile-verified
//
#include <hip/hip_runtime.h>
#include <hip/hip_bf16.h>
#include <math.h>

// Problem dims (fixed by the reference)
#define B_   4
#define Q_   128
#define K_   1024
#define H_   256
#define QS_  256
#define DV_  256

typedef __attribute__((ext_vector_type(2))) float v2f;
typedef __attribute__((ext_vector_type(8))) float v8f;

static __device__ __forceinline__ v8f wmma_f32_16x16x4(v2f a, v2f b, v8f c) {
    // 8 args: (neg_a, A, neg_b, B, c_mod, C, reuse_a, reuse_b)
    return __builtin_amdgcn_wmma_f32_16x16x4_f32(false, a, false, b,
                                                 (short)0, c, false, false);
}

// ---------------------------------------------------------------------------
// fp32 WMMA GEMM, register-blocked 2x2 tiles per wave (32x32 C per wave):
//   C[M,N] = A[M,KD] * Bm[KD,N], all row-major, N/KD compile-time constants.
// 4 wmma per k-step sharing 2 A-fragments and 2 B-fragments -> 2 loads/wmma.
// grid = (M/32, N/32, batch); blockDim = 32 (EXEC all-ones, as WMMA requires).
// ---------------------------------------------------------------------------
template <int N, int KD>
__global__ void wmma_gemm_f32_kernel(const float* __restrict__ A,
                                     const float* __restrict__ Bm,
                                     float* __restrict__ C,
                                     int strideA, int strideB, int strideC)
{
    const int lane = threadIdx.x & 31;
    const int tm   = blockIdx.x * 2;   // first of 2 M-tiles
    const int tn   = blockIdx.y * 2;   // first of 2 N-tiles
    const int b    = blockIdx.z;

    const float* __restrict__ Ab = A  + b * strideA;
    const float* __restrict__ Bb = Bm + b * strideB;
    float* __restrict__       Cb = C  + b * strideC;

    const int l15   = lane & 15;
    const int khalf = (lane >> 4) * 2;          // 0 (lanes 0-15) or 2 (16-31)

    const int arow0 = (tm + 0) * 16 + l15;      // A rows held by this lane
    const int arow1 = (tm + 1) * 16 + l15;
    const int bcol0 = (tn + 0) * 16 + l15;      // B cols held by this lane
    const int bcol1 = (tn + 1) * 16 + l15;

    v8f acc00 = {}, acc01 = {}, acc10 = {}, acc11 = {};

#pragma unroll 4
    for (int k0 = 0; k0 < KD; k0 += 4) {
        v2f a0, a1, b0, b1;
        const float* ap0 = Ab + arow0 * KD + (k0 + khalf);   // b64 per lane
        const float* ap1 = Ab + arow1 * KD + (k0 + khalf);
        a0.x = ap0[0]; a0.y = ap0[1];
        a1.x = ap1[0]; a1.y = ap1[1];
        const float* bp = Bb + (k0 + khalf) * N;
        b0.x = bp[bcol0]; b0.y = bp[N + bcol0];
        b1.x = bp[bcol1]; b1.y = bp[N + bcol1];

        acc00 = wmma_f32_16x16x4(a0, b0, acc00);
        acc01 = wmma_f32_16x16x4(a0, b1, acc01);
        acc10 = wmma_f32_16x16x4(a1, b0, acc10);
        acc11 = wmma_f32_16x16x4(a1, b1, acc11);
    }

    // D layout: VGPR r -> M = 16*tile + r + (lane>=16 ? 8 : 0), N = 16*tile + l15
    const int rb0 = (tm + 0) * 16 + ((lane >> 4) * 8);
    const int rb1 = (tm + 1) * 16 + ((lane >> 4) * 8);
#pragma unroll
    for (int r = 0; r < 8; ++r) {
        Cb[(rb0 + r) * N + bcol0] = acc00[r];
        Cb[(rb0 + r) * N + bcol1] = acc01[r];
        Cb[(rb1 + r) * N + bcol0] = acc10[r];
        Cb[(rb1 + r) * N + bcol1] = acc11[r];
    }
}

// ---------------------------------------------------------------------------
// Fused scores + mask + softmax. One 256-thread block per (b,q) row.
// score(k) = sum_h tanh(qrow[h] + krow[h]) * Wv[h]; tanh via exp+rcp trans ops.
// k-rows streamed as float4 (global_load_b128); qrow/Wv cached in LDS.
// ---------------------------------------------------------------------------
__global__ void scores_softmax_kernel(const float* __restrict__ qp,   // [B*Q, H]
                                      const float* __restrict__ kp,   // [B*K, H]
                                      const float* __restrict__ Wv,   // [H]
                                      const long long* __restrict__ valid_lens, // [B]
                                      float* __restrict__ attn)       // [B*Q, K]
{
    const int bq  = blockIdx.x;        // 0 .. B*Q-1
    const int b   = bq / Q_;
    const int tid = threadIdx.x;       // 0 .. 255

    __shared__ float qrow[H_];
    __shared__ float wv[H_];
    __shared__ float red[256];

    qrow[tid] = qp[bq * H_ + tid];
    wv[tid]   = Wv[tid];
    __syncthreads();

    const int vlen = (int)valid_lens[b];
    const float4* __restrict__ q4 = reinterpret_cast<const float4*>(qrow);
    const float4* __restrict__ w4 = reinterpret_cast<const float4*>(wv);

    float sc[K_ / 256];
    float lmax = -3.0e38f;
#pragma unroll
    for (int i = 0; i < K_ / 256; ++i) {
        const int k = tid + i * 256;
        const float4* __restrict__ k4 =
            reinterpret_cast<const float4*>(kp + (b * K_ + k) * H_);
        float s = 0.0f;
#pragma unroll 2
        for (int h4 = 0; h4 < H_ / 4; ++h4) {
            const float4 kv = k4[h4];
            const float4 qv = q4[h4];
            const float4 wvv = w4[h4];
            // tanh(x) = (e-1)*rcp(e+1), e = exp(2x): v_exp_f32 + v_rcp_f32
            float e, t;
            e = __expf(2.0f * (qv.x + kv.x));
            t = (e - 1.0f) * __builtin_amdgcn_rcpf(e + 1.0f);
            s = __builtin_fmaf(t, wvv.x, s);
            e = __expf(2.0f * (qv.y + kv.y));
            t = (e - 1.0f) * __builtin_amdgcn_rcpf(e + 1.0f);
            s = __builtin_fmaf(t, wvv.y, s);
            e = __expf(2.0f * (qv.z + kv.z));
            t = (e - 1.0f) * __builtin_amdgcn_rcpf(e + 1.0f);
            s = __builtin_fmaf(t, wvv.z, s);
            e = __expf(2.0f * (qv.w + kv.w));
            t = (e - 1.0f) * __builtin_amdgcn_rcpf(e + 1.0f);
            s = __builtin_fmaf(t, wvv.w, s);
        }
        if (k >= vlen) s = -1.0e6f;    // sequence_mask semantics
        sc[i] = s;
        lmax  = fmaxf(lmax, s);
    }

    // Block max reduction
    red[tid] = lmax;
    __syncthreads();
    for (int off = 128; off > 0; off >>= 1) {
        if (tid < off) red[tid] = fmaxf(red[tid], red[tid + off]);
        __syncthreads();
    }
    const float m = red[0];
    __syncthreads();

    // exp + block sum reduction
    float lsum = 0.0f;
#pragma unroll
    for (int i = 0; i < K_ / 256; ++i) {
        sc[i] = __expf(sc[i] - m);
        lsum += sc[i];
    }
    red[tid] = lsum;
    __syncthreads();
    for (int off = 128; off > 0; off >>= 1) {
        if (tid < off) red[tid] += red[tid + off];
        __syncthreads();
    }
    const float inv = __builtin_amdgcn_rcpf(red[0]);

#pragma unroll
    for (int i = 0; i < K_ / 256; ++i) {
        attn[bq * K_ + tid + i * 256] = sc[i] * inv;
    }
}

// ---------------------------------------------------------------------------
// Launch: q-proj GEMM, k-proj GEMM, fused scores/softmax, attn@V GEMM.
// Workspace: qp (512x256 f32) | kp (4096x256 f32) | attn (512x1024 f32) ~ 6.5MB
// ---------------------------------------------------------------------------
extern "C" void kernel_launch(void* const* d_in, const int* in_sizes, int n_in,
                              void* d_out, int out_size, void* d_ws, size_t ws_size,
                              hipStream_t stream)
{
    const float*     queries    = (const float*)d_in[0];     // [B,Q,QS]
    const float*     keys       = (const float*)d_in[1];     // [B,K,KS]
    const float*     values     = (const float*)d_in[2];     // [B,K,DV]
    const long long* valid_lens = (const long long*)d_in[3]; // [B] int64
    const float*     Wq         = (const float*)d_in[4];     // [QS,H]
    const float*     Wk         = (const float*)d_in[5];     // [KS,H]
    const float*     Wv         = (const float*)d_in[6];     // [H]
    float*           out        = (float*)d_out;             // [B,Q,DV]

    float* qp   = (float*)d_ws;                      // [B*Q, H]
    float* kp   = qp + (size_t)B_ * Q_ * H_;         // [B*K, H]
    float* attn = kp + (size_t)B_ * K_ * H_;         // [B*Q, K]

    const dim3 wave(32, 1, 1);

    // q = queries @ Wq : [512,256] x [256,256], 32x32 C per wave
    wmma_gemm_f32_kernel<H_, QS_>
        <<<dim3((B_ * Q_) / 32, H_ / 32, 1), wave, 0, stream>>>(
            queries, Wq, qp, 0, 0, 0);

    // k = keys @ Wk : [4096,256] x [256,256]
    wmma_gemm_f32_kernel<H_, QS_>
        <<<dim3((B_ * K_) / 32, H_ / 32, 1), wave, 0, stream>>>(
            keys, Wk, kp, 0, 0, 0);

    // fused tanh-scores + mask + softmax : one block per (b,q)
    scores_softmax_kernel<<<dim3(B_ * Q_, 1, 1), dim3(256, 1, 1), 0, stream>>>(
        qp, kp, Wv, valid_lens, attn);

    // out = attn @ values (batched) : [128,1024] x [1024,256] per batch
    wmma_gemm_f32_kernel<DV_, K_>
        <<<dim3(Q_ / 32, DV_ / 32, B_), wave, 0, stream>>>(
            attn, values, out, Q_ * K_, K_ * DV_, Q_ * DV_);
}